// U3Layer_47055661695345
// MI455X (gfx1250) — compile-verified
//
#include <hip/hip_runtime.h>
#include <math.h>

// Problem geometry
#define DIMN    4096          // 2^12 rows
#define MCOLS   4096          // state-vector columns
#define CHUNKC  2048          // column chunk: 128MB complex intermediate < 192MB L2
#define TR      64            // tile rows (one 6-qubit butterfly group)
#define TC      64            // tile cols
#define NT      256           // threads per block (8 wave32)
#define ELEMS   (TR * TC)     // 4096 complex elements per tile
#define ITERS   (ELEMS / NT)  // 16

// Temporal hints (gfx12 CPol: TH in bits [2:0])
#define TH_RT 0   // regular temporal (keep in L2) — intermediate store
#define TH_NT 1   // non-temporal (stream, don't pollute) — input loads / final stores
#define TH_LU 3   // last-use (read + release) — intermediate loads

// Vector-int types matching the async builtin signatures
typedef int v2i __attribute__((vector_size(8)));
typedef int v4i __attribute__((vector_size(16)));
#define AS1 __attribute__((address_space(1)))
#define AS3 __attribute__((address_space(3)))

// ---------------- CDNA5 async global<->LDS path (probe-guarded) ----------------
#if defined(__HIP_DEVICE_COMPILE__) && defined(__gfx1250__)
  #if __has_builtin(__builtin_amdgcn_global_load_async_to_lds_b64)
    #define A_LD64(g, l, cp) __builtin_amdgcn_global_load_async_to_lds_b64( \
        (AS1 v2i*)(g), (AS3 v2i*)(l), 0, (cp))
  #endif
  #if __has_builtin(__builtin_amdgcn_global_load_async_to_lds_b128)
    #define A_LD128(g, l, cp) __builtin_amdgcn_global_load_async_to_lds_b128( \
        (AS1 v4i*)(g), (AS3 v4i*)(l), 0, (cp))
  #endif
  #if __has_builtin(__builtin_amdgcn_global_store_async_from_lds_b128)
    #define A_ST128(g, l, cp) __builtin_amdgcn_global_store_async_from_lds_b128( \
        (AS1 v4i*)(g), (AS3 v4i*)(l), 0, (cp))
  #endif
  #if __has_builtin(__builtin_amdgcn_s_wait_asynccnt)
    #define A_WAIT0() __builtin_amdgcn_s_wait_asynccnt(0)
  #else
    #define A_WAIT0() asm volatile("s_wait_asynccnt 0" ::: "memory")
  #endif
#endif
// Fallbacks (host pass, or builtin not present) — synchronous equivalents.
#ifndef A_LD64
  #define A_LD64(g, l, cp)  (*(double*)(l) = *(const double*)(g))
#endif
#ifndef A_LD128
  #define A_LD128(g, l, cp) (*(double2*)(l) = *(const double2*)(g))
#endif
#ifndef A_ST128
  #define A_ST128(g, l, cp) (*(double2*)(g) = *(const double2*)(l))
#endif
#ifndef A_WAIT0
  #define A_WAIT0() ((void)0)
#endif

// U3 gate (2x2 complex) from 3 Euler angles, matching the reference:
//   [[cos(t/2),              -e^{i p2} sin(t/2)     ],
//    [e^{i p1} sin(t/2),      e^{i (p1+p2)} cos(t/2)]]
// gates layout: [u00r,u00i, u01r,u01i, u10r,u10i, u11r,u11i] per qubit.
__device__ __forceinline__ void compute_gates(const double* __restrict__ thetas,
                                              int qbase, double* gates) {
  const int tid = threadIdx.x;
  if (tid < 6) {
    const int q = qbase + tid;
    const double th = thetas[q * 3 + 0] * 0.5;
    const double p1 = thetas[q * 3 + 1];
    const double p2 = thetas[q * 3 + 2];
    const double c = cos(th), s = sin(th);
    double* g = gates + tid * 8;
    g[0] = c;               g[1] = 0.0;
    g[2] = -cos(p2) * s;    g[3] = -sin(p2) * s;
    g[4] =  cos(p1) * s;    g[5] =  sin(p1) * s;
    g[6] =  cos(p1 + p2) * c; g[7] = sin(p1 + p2) * c;
  }
}

// Six butterfly stages (strides 32,16,8,4,2,1 in tile-row space) over a
// 64x64 complex tile resident in LDS. Caller must __syncthreads() before.
__device__ __forceinline__ void butterfly6(double2* tile, const double* gates) {
  const int tid = threadIdx.x;
  #pragma unroll
  for (int j = 0; j < 6; ++j) {
    const int logs = 5 - j;          // stride = 32 >> j
    const int s = 1 << logs;
    const double g0r = gates[j*8+0], g0i = gates[j*8+1];
    const double g1r = gates[j*8+2], g1i = gates[j*8+3];
    const double g2r = gates[j*8+4], g2i = gates[j*8+5];
    const double g3r = gates[j*8+6], g3i = gates[j*8+7];
    #pragma unroll
    for (int it = 0; it < (TR / 2) * TC / NT; ++it) {   // 8 pairs/thread
      const int idx = tid + it * NT;                    // 0..2047
      const int col = idx & (TC - 1);
      const int pr  = idx >> 6;                         // 0..31
      const int ra  = ((pr >> logs) << (logs + 1)) | (pr & (s - 1));
      const int rb  = ra + s;
      const double2 a = tile[ra * TC + col];
      const double2 b = tile[rb * TC + col];
      double2 na, nb;
      na.x = g0r * a.x - g0i * a.y + g1r * b.x - g1i * b.y;
      na.y = g0r * a.y + g0i * a.x + g1r * b.y + g1i * b.x;
      nb.x = g2r * a.x - g2i * a.y + g3r * b.x - g3i * b.y;
      nb.y = g2r * a.y + g2i * a.x + g3r * b.y + g3i * b.x;
      tile[ra * TC + col] = na;
      tile[rb * TC + col] = nb;
    }
    __syncthreads();
  }
}

// Pass 1: qubits 6..11 (row strides 32..1) on 64-row-aligned groups, for one
// column chunk. Reads planar fp64 re/im (NT: streamed), writes interleaved
// complex128 intermediate (RT: stays dirty in L2 for pass 2 of this chunk).
__global__ __launch_bounds__(NT) void u3_low6_kernel(
    const double* __restrict__ thetas,
    const double* __restrict__ re,
    const double* __restrict__ im,
    double2* __restrict__ out, int col0) {
  __shared__ double2 tile[ELEMS];
  __shared__ double gates[48];
  const int tid = threadIdx.x;
  const int bc = blockIdx.x & (CHUNKC / TC - 1);   // 0..31
  const int br = blockIdx.x >> 5;                  // 0..63
  const long r0 = (long)br * TR;
  const long c0 = (long)col0 + (long)bc * TC;

  #pragma unroll
  for (int it = 0; it < ITERS; ++it) {
    const int e = tid + it * NT;
    const int r = e >> 6, c = e & (TC - 1);
    const long gidx = (r0 + r) * (long)MCOLS + (c0 + c);
    A_LD64(re + gidx, &tile[e].x, TH_NT);   // async DMA, streamed
    A_LD64(im + gidx, &tile[e].y, TH_NT);
  }
  compute_gates(thetas, /*qbase=*/6, gates);  // overlaps with async loads
  A_WAIT0();
  __syncthreads();

  butterfly6(tile, gates);

  #pragma unroll
  for (int it = 0; it < ITERS; ++it) {
    const int e = tid + it * NT;
    const int r = e >> 6, c = e & (TC - 1);
    const long gidx = (r0 + r) * (long)MCOLS + (c0 + c);
    A_ST128(out + gidx, &tile[e], TH_RT);   // keep intermediate in L2
  }
  A_WAIT0();
}

// Pass 2: qubits 0..5 (row strides 2048..64). Rows {g*64 + rlow}, all g, for
// one column chunk. In-place on the interleaved buffer; per-block read set ==
// write set. Loads are last-use (intermediate dies), stores are streamed.
__global__ __launch_bounds__(NT) void u3_high6_kernel(
    const double* __restrict__ thetas,
    double2* __restrict__ data, int col0) {
  __shared__ double2 tile[ELEMS];
  __shared__ double gates[48];
  const int tid = threadIdx.x;
  const int rlow = blockIdx.x & (TR - 1);          // 0..63
  const int bc = blockIdx.x >> 6;                  // 0..31
  const long c0 = (long)col0 + (long)bc * TC;

  #pragma unroll
  for (int it = 0; it < ITERS; ++it) {
    const int e = tid + it * NT;
    const int g = e >> 6, c = e & (TC - 1);
    const long row = (long)g * TR + rlow;
    const long gidx = row * (long)MCOLS + (c0 + c);
    A_LD128(data + gidx, &tile[e], TH_LU);  // last use of intermediate
  }
  compute_gates(thetas, /*qbase=*/0, gates);
  A_WAIT0();
  __syncthreads();

  butterfly6(tile, gates);

  #pragma unroll
  for (int it = 0; it < ITERS; ++it) {
    const int e = tid + it * NT;
    const int g = e >> 6, c = e & (TC - 1);
    const long row = (long)g * TR + rlow;
    const long gidx = row * (long)MCOLS + (c0 + c);
    A_ST128(data + gidx, &tile[e], TH_NT);  // stream final result out
  }
  A_WAIT0();
}

extern "C" void kernel_launch(void* const* d_in, const int* in_sizes, int n_in,
                              void* d_out, int out_size, void* d_ws, size_t ws_size,
                              hipStream_t stream) {
  (void)in_sizes; (void)n_in; (void)out_size; (void)d_ws; (void)ws_size;
  const double* thetas = (const double*)d_in[0];   // [12,3] fp64
  const double* re     = (const double*)d_in[1];   // [4096,4096] fp64
  const double* im     = (const double*)d_in[2];   // [4096,4096] fp64
  double2* out = (double2*)d_out;                  // complex128 interleaved

  const int nblocks1 = (DIMN / TR) * (CHUNKC / TC);  // 2048
  const int nblocks2 = TR * (CHUNKC / TC);           // 2048
  // Process column chunks pass1->pass2 so the 128MB intermediate stays in the
  // 192MB L2 and never round-trips HBM.
  for (int col0 = 0; col0 < MCOLS; col0 += CHUNKC) {
    u3_low6_kernel<<<nblocks1, NT, 0, stream>>>(thetas, re, im, out, col0);
    u3_high6_kernel<<<nblocks2, NT, 0, stream>>>(thetas, out, col0);
  }
}